// LocationMemoryBank_67800353734815
// MI455X (gfx1250) — compile-verified
//
#include <hip/hip_runtime.h>

typedef __attribute__((ext_vector_type(2))) float v2f;
typedef __attribute__((ext_vector_type(8))) float v8f;

#define K_RECENT 8
#define D_FEAT   512   // reference feature dim (fixed)
#define M_SLOTS  20    // reference memory slots (fixed)
#define WAVES_PER_BLOCK 8

// One wave32 per query. out[b,:] = sum_{p<k} softmax(arange(k))[p] * feats[loc, start+p, :]
//
// WMMA mapping (V_WMMA_F32_16X16X4_F32, f32 in/out => bit-accurate fp32 path):
//   A (16x4): every row = (w0..w3) [then (w4..w7) for the accumulate step]
//   B (4x16): 4 slot rows x 16 consecutive feature columns
//   C (16x16): every row = the 16 outputs for this feature chunk
// A layout: VGPR0 = K0|K2 per lane half, VGPR1 = K1|K3; B mirrors.
// D: acc[0] holds out[d0 + (lane&15)] in BOTH lane halves (M=0 and M=8 rows
// identical), so a full-wave store writes duplicate-but-identical data =>
// uniform control flow everywhere (WMMA requires EXEC all-ones anyway).
__global__ __launch_bounds__(32 * WAVES_PER_BLOCK) void lmb_wmma_kernel(
    const float* __restrict__ feats,   // [L, M, D]
    const int*   __restrict__ counts,  // [L]
    const int*   __restrict__ loc_idx, // [B]
    float*       __restrict__ out,     // [B, D]
    int B) {
  const int lane = threadIdx.x & 31;
  // Wave-uniform query id in an SGPR -> scalar index loads, SALU weight math.
  const int b = __builtin_amdgcn_readfirstlane(
      (int)(blockIdx.x * WAVES_PER_BLOCK + (threadIdx.x >> 5)));
  if (b >= B) return;  // uniform branch

  const int loc   = __builtin_amdgcn_readfirstlane(loc_idx[b]);
  const int c     = counts[loc];                 // scalar load
  const int k     = c < K_RECENT ? c : K_RECENT;
  const int start = c - k;  // start+7 <= 19 always (k==8 -> start<=12; k<8 -> start==0)

  // softmax(arange(k)): integer logits -> constant e^p table, no expf.
  const float E[K_RECENT] = {1.0f, 2.718281828459045f, 7.389056098930650f,
                             20.085536923187668f, 54.598150033144236f,
                             148.41315910257660f, 403.42879349273512f,
                             1096.6331584284585f};
  float denom = 0.0f;
#pragma unroll
  for (int q = 0; q < K_RECENT; ++q) denom += (q < k) ? E[q] : 0.0f;
  const float rdenom = (k > 0) ? 1.0f / denom : 0.0f;  // k==0 -> zero output
  float w[K_RECENT];
#pragma unroll
  for (int p = 0; p < K_RECENT; ++p) w[p] = (p < k) ? E[p] * rdenom : 0.0f;

  const bool hi = lane >= 16;
  // A matrices: identical weight row replicated 16x; lane halves pick K pairs.
  v2f a_lo, a_hi;
  a_lo[0] = hi ? w[2] : w[0];
  a_lo[1] = hi ? w[3] : w[1];
  a_hi[0] = hi ? w[6] : w[4];
  a_hi[1] = hi ? w[7] : w[5];

  // Per-lane element offsets off a wave-uniform base (compile-time strides).
  const int n    = lane & 15;
  const int rsel = hi ? 2 : 0;
  const float* base = feats + ((size_t)loc * M_SLOTS + start) * D_FEAT;
  const float* pl0 = base + (rsel + 0) * D_FEAT + n;  // B_lo VGPR0: rows {0,2}
  const float* pl1 = base + (rsel + 1) * D_FEAT + n;  // B_lo VGPR1: rows {1,3}
  const float* ph0 = base + (rsel + 4) * D_FEAT + n;  // B_hi VGPR0: rows {4,6}
  const float* ph1 = base + (rsel + 5) * D_FEAT + n;  // B_hi VGPR1: rows {5,7}
  float*       op  = out + (size_t)b * D_FEAT + n;

#pragma unroll 4
  for (int d0 = 0; d0 < D_FEAT; d0 += 16) {
    v2f bl, bh;
    bl[0] = pl0[d0];
    bl[1] = pl1[d0];
    bh[0] = ph0[d0];
    bh[1] = ph1[d0];

    v8f acc = {};
    acc = __builtin_amdgcn_wmma_f32_16x16x4_f32(false, a_lo, false, bl,
                                                (short)0, acc, false, false);
    acc = __builtin_amdgcn_wmma_f32_16x16x4_f32(false, a_hi, false, bh,
                                                (short)0, acc, false, false);

    // Full-wave store; lane halves write identical data to identical
    // addresses (64B of unique data per chunk). EXEC stays all-ones.
    op[d0] = acc[0];
  }
}

extern "C" void kernel_launch(void* const* d_in, const int* in_sizes, int n_in,
                              void* d_out, int out_size, void* d_ws, size_t ws_size,
                              hipStream_t stream) {
  const float* feats   = (const float*)d_in[0];  // [L, M, D] f32
  const int*   counts  = (const int*)d_in[1];    // [L] i32
  const int*   loc_idx = (const int*)d_in[2];    // [B] i32

  const int Bq = in_sizes[2];
  float* out = (float*)d_out;

  dim3 block(32 * WAVES_PER_BLOCK);
  dim3 grid((Bq + WAVES_PER_BLOCK - 1) / WAVES_PER_BLOCK);
  lmb_wmma_kernel<<<grid, block, 0, stream>>>(feats, counts, loc_idx, out, Bq);
}